// AttentionGraphEncoder_45320494908029
// MI455X (gfx1250) — compile-verified
//
#include <hip/hip_runtime.h>
#include <hip/hip_bf16.h>
#include <math.h>

#define BSZ  256
#define NN   2048
#define EMBD 128
#define NORMC 0.08838834764831845f   // 1/sqrt(128)

typedef float v2f __attribute__((ext_vector_type(2)));
typedef float v8f __attribute__((ext_vector_type(8)));

// ---- workspace layout (float offsets) ----
#define OFF_QIN   0                         // [256,384]
#define OFF_Q     (BSZ*384)                 // [256,128]
#define OFF_K0    (OFF_Q  + BSZ*EMBD)       // [256,128]
#define OFF_V0    (OFF_K0 + BSZ*EMBD)       // [256,128]
#define OFF_FOLD  (OFF_V0 + BSZ*EMBD)       // 1792 floats
// fold sublayout
#define F_MK   0      // [3,128]  W_node @ w_k_bot
#define F_MDK  384    // [2,128]  W_depot @ w_k_bot
#define F_BK   640    // [128]    b_node @ w_k_bot
#define F_BDK  768    // [128]    b_depot @ w_k_bot
#define F_MV   896    // [3,128]
#define F_MDV  1280   // [2,128]
#define F_BV   1536   // [128]
#define F_BDV  1664   // [128]

// ---------------------------------------------------------------------------
// Kernel A: fold the rank-3 node embedding through the bottom halves of w_k/w_v
// ---------------------------------------------------------------------------
__global__ void fold_kernel(const float* __restrict__ W_node, const float* __restrict__ b_node,
                            const float* __restrict__ W_depot, const float* __restrict__ b_depot,
                            const float* __restrict__ w_k, const float* __restrict__ w_v,
                            float* __restrict__ fold) {
    int t = threadIdx.x;  // 0..127 (output column)
    float mk0 = 0.f, mk1 = 0.f, mk2 = 0.f, mdk0 = 0.f, mdk1 = 0.f, bk = 0.f, bdk = 0.f;
    float mv0 = 0.f, mv1 = 0.f, mv2 = 0.f, mdv0 = 0.f, mdv1 = 0.f, bv = 0.f, bdv = 0.f;
    for (int e = 0; e < EMBD; ++e) {
        float wk = w_k[(EMBD + e) * EMBD + t];   // bottom half of w_k
        float wv = w_v[(EMBD + e) * EMBD + t];   // bottom half of w_v
        float n0 = W_node[0 * EMBD + e], n1 = W_node[1 * EMBD + e], n2 = W_node[2 * EMBD + e];
        float d0 = W_depot[0 * EMBD + e], d1 = W_depot[1 * EMBD + e];
        float bn = b_node[e], bd = b_depot[e];
        mk0 += n0 * wk;  mk1 += n1 * wk;  mk2 += n2 * wk;
        mdk0 += d0 * wk; mdk1 += d1 * wk;
        bk  += bn * wk;  bdk += bd * wk;
        mv0 += n0 * wv;  mv1 += n1 * wv;  mv2 += n2 * wv;
        mdv0 += d0 * wv; mdv1 += d1 * wv;
        bv  += bn * wv;  bdv += bd * wv;
    }
    fold[F_MK  + 0 * EMBD + t] = mk0;
    fold[F_MK  + 1 * EMBD + t] = mk1;
    fold[F_MK  + 2 * EMBD + t] = mk2;
    fold[F_MDK + 0 * EMBD + t] = mdk0;
    fold[F_MDK + 1 * EMBD + t] = mdk1;
    fold[F_BK  + t] = bk;
    fold[F_BDK + t] = bdk;
    fold[F_MV  + 0 * EMBD + t] = mv0;
    fold[F_MV  + 1 * EMBD + t] = mv1;
    fold[F_MV  + 2 * EMBD + t] = mv2;
    fold[F_MDV + 0 * EMBD + t] = mdv0;
    fold[F_MDV + 1 * EMBD + t] = mdv1;
    fold[F_BV  + t] = bv;
    fold[F_BDV + t] = bdv;
}

// ---------------------------------------------------------------------------
// Kernel B: gather curr/next node embeddings + state embedding -> Qin[256,384]
// ---------------------------------------------------------------------------
__device__ __forceinline__ float node_embed(const float* __restrict__ nf,
                                            const float* __restrict__ W_node,
                                            const float* __restrict__ b_node,
                                            const float* __restrict__ W_depot,
                                            const float* __restrict__ b_depot,
                                            long base, int id, int t) {
    const float* fr = nf + base + (long)id * 3;
    if (id == 0)  // depot: first 2 coords through W_depot
        return fr[0] * W_depot[t] + fr[1] * W_depot[EMBD + t] + b_depot[t];
    return fr[0] * W_node[t] + fr[1] * W_node[EMBD + t] + fr[2] * W_node[2 * EMBD + t] + b_node[t];
}

__global__ void embed_kernel(const float* __restrict__ nf, const float* __restrict__ state,
                             const float* __restrict__ W_node, const float* __restrict__ b_node,
                             const float* __restrict__ W_depot, const float* __restrict__ b_depot,
                             const float* __restrict__ W_state, const float* __restrict__ b_state,
                             const int* __restrict__ curr_id, const int* __restrict__ next_id,
                             float* __restrict__ Qin) {
    int b = blockIdx.x;
    int t = threadIdx.x;  // 0..127
    long base = (long)b * NN * 3;
    float ce = node_embed(nf, W_node, b_node, W_depot, b_depot, base, curr_id[b], t);
    float ne = node_embed(nf, W_node, b_node, W_depot, b_depot, base, next_id[b], t);
    float se = b_state[t];
    #pragma unroll
    for (int s = 0; s < 4; ++s) se += state[b * 4 + s] * W_state[s * EMBD + t];
    Qin[(long)b * 384 + t]       = ce;   // curr_emb (also A-matrix for k0/v0 GEMMs)
    Qin[(long)b * 384 + 128 + t] = ne;
    Qin[(long)b * 384 + 256 + t] = se;
}

// ---------------------------------------------------------------------------
// Kernel C: fp32 WMMA GEMM  C[256,128] = A[256,K] @ B[K,128]
// one wave per 16x16 tile, V_WMMA_F32_16X16X4_F32, K stepped by 4
// ---------------------------------------------------------------------------
__global__ void gemm_wmma(const float* __restrict__ A, int lda,
                          const float* __restrict__ Bm, int ldb,
                          int K, float* __restrict__ C, int ldc) {
    int wave = threadIdx.x >> 5;          // 4 waves / block
    int lane = threadIdx.x & 31;
    int tile = blockIdx.x * 4 + wave;     // 128 tiles total (16 x 8)
    int m0 = (tile >> 3) * 16;
    int n0 = (tile & 7) * 16;
    int half = lane >> 4;                 // 0: lanes 0-15, 1: lanes 16-31
    int lrow = lane & 15;

    v8f acc = {};
    // A frag: lanes0-15 hold K={k,k+1}, lanes16-31 hold K={k+2,k+3} of row m0+lrow
    const float* arow = A + (long)(m0 + lrow) * lda + 2 * half;
    // B frag: b.x row = k+2*half, b.y row = k+1+2*half, col n0+lrow
    const float* bcol = Bm + (long)(2 * half) * ldb + n0 + lrow;
    for (int k = 0; k < K; k += 4) {
        v2f a; a.x = arow[k];           a.y = arow[k + 1];
        v2f b; b.x = bcol[(long)k * ldb]; b.y = bcol[(long)(k + 1) * ldb];
        acc = __builtin_amdgcn_wmma_f32_16x16x4_f32(
            /*neg_a=*/false, a, /*neg_b=*/false, b,
            /*c_mod=*/(short)0, acc, /*reuse_a=*/false, /*reuse_b=*/false);
    }
    #pragma unroll
    for (int r = 0; r < 8; ++r)  // C VGPR r: lanes0-15 -> row r, lanes16-31 -> row r+8
        C[(long)(m0 + r + 8 * half) * ldc + n0 + lrow] = acc[r];
}

// ---------------------------------------------------------------------------
// Kernel D: fused attention per batch. One block (256 thr) per b.
// logit[n] = NORM*(feats[n].mkq + cK), online masked softmax, then
// h = (Sum attn*feats)@Mv / S + bias/depot terms + v0[b]
// ---------------------------------------------------------------------------
__global__ void attn_kernel(const float* __restrict__ node_feats,
                            const unsigned char* __restrict__ mask,
                            const float* __restrict__ qbuf,
                            const float* __restrict__ k0buf,
                            const float* __restrict__ v0buf,
                            const float* __restrict__ fold,
                            float* __restrict__ out) {
    int b = blockIdx.x;
    int t = threadIdx.x;  // 0..255

    __shared__ float part[7 * EMBD];
    __shared__ float scal[7];
    __shared__ float d2sh[2];
    __shared__ float red[6 * 256];

    const float* Mk    = fold + F_MK;
    const float* Mdk   = fold + F_MDK;
    const float* bkrow = fold + F_BK;
    const float* bdkrow= fold + F_BDK;
    const float* Mv    = fold + F_MV;
    const float* Mdv   = fold + F_MDV;
    const float* bvrow = fold + F_BV;
    const float* bdvrow= fold + F_BDV;

    const float* frow_base = node_feats + (long)b * NN * 3;

    // ---- per-batch scalars: mkq = Mk@q, mdq = Mdk@q, cK = q.(bk+k0), cKd = q.(bdk+k0)
    if (t < EMBD) {
        float qv  = qbuf[b * EMBD + t];
        float k0v = k0buf[b * EMBD + t];
        part[0 * EMBD + t] = Mk[0 * EMBD + t] * qv;
        part[1 * EMBD + t] = Mk[1 * EMBD + t] * qv;
        part[2 * EMBD + t] = Mk[2 * EMBD + t] * qv;
        part[3 * EMBD + t] = Mdk[0 * EMBD + t] * qv;
        part[4 * EMBD + t] = Mdk[1 * EMBD + t] * qv;
        part[5 * EMBD + t] = (bkrow[t]  + k0v) * qv;
        part[6 * EMBD + t] = (bdkrow[t] + k0v) * qv;
    }
    if (t == 0) { d2sh[0] = frow_base[0]; d2sh[1] = frow_base[1]; }
    __syncthreads();
    if (t < 7) {
        float acc = 0.f;
        for (int e = 0; e < EMBD; ++e) acc += part[t * EMBD + e];
        scal[t] = acc;
    }
    __syncthreads();
    float mkq0 = scal[0], mkq1 = scal[1], mkq2 = scal[2];
    float mdq0 = scal[3], mdq1 = scal[4];
    float cK = scal[5], cKd = scal[6];

    // ---- streaming pass: online masked softmax with weighted-feature accum
    const unsigned char* mrow = mask + (long)b * NN;
    float m = -INFINITY, s = 0.f, f0 = 0.f, f1 = 0.f, f2 = 0.f, dpt = 0.f;
    for (int n = t; n < NN; n += 256) {
        if (!mrow[n]) continue;
        const float* fr = frow_base + (long)n * 3;
        float x = fr[0], y = fr[1];
        float l;
        if (n == 0) {
            l = NORMC * (x * mdq0 + y * mdq1 + cKd);
        } else {
            float z = fr[2];
            l = NORMC * (x * mkq0 + y * mkq1 + z * mkq2 + cK);
            // fall-through uses z below via recompute guard
        }
        float nm = fmaxf(m, l);
        float sc = (s > 0.f) ? __expf(m - nm) : 0.f;
        float e  = __expf(l - nm);
        s = s * sc + e;
        f0 *= sc; f1 *= sc; f2 *= sc; dpt *= sc;
        if (n == 0) {
            dpt += e;
        } else {
            f0 += e * x; f1 += e * y; f2 += e * fr[2];
        }
        m = nm;
    }
    red[0 * 256 + t] = m;  red[1 * 256 + t] = s;
    red[2 * 256 + t] = f0; red[3 * 256 + t] = f1;
    red[4 * 256 + t] = f2; red[5 * 256 + t] = dpt;

    for (int off = 128; off > 0; off >>= 1) {
        __syncthreads();
        if (t < off) {
            float m2 = red[0 * 256 + t + off], s2 = red[1 * 256 + t + off];
            if (s2 > 0.f) {
                float m1 = red[0 * 256 + t], s1 = red[1 * 256 + t];
                float nm = fmaxf(m1, m2);
                float sc1 = (s1 > 0.f) ? __expf(m1 - nm) : 0.f;
                float sc2 = __expf(m2 - nm);
                red[0 * 256 + t] = nm;
                red[1 * 256 + t] = s1 * sc1 + s2 * sc2;
                red[2 * 256 + t] = red[2 * 256 + t] * sc1 + red[2 * 256 + t + off] * sc2;
                red[3 * 256 + t] = red[3 * 256 + t] * sc1 + red[3 * 256 + t + off] * sc2;
                red[4 * 256 + t] = red[4 * 256 + t] * sc1 + red[4 * 256 + t + off] * sc2;
                red[5 * 256 + t] = red[5 * 256 + t] * sc1 + red[5 * 256 + t + off] * sc2;
            }
        }
    }
    __syncthreads();

    if (t < EMBD) {
        float S  = red[1 * 256];
        float F0 = red[2 * 256], F1 = red[3 * 256], F2 = red[4 * 256];
        float D  = red[5 * 256];
        float invS = 1.0f / S;
        float a0   = D * invS;            // depot attention weight
        float rest = (S - D) * invS;      // sum of non-depot attention
        float h = (F0 * Mv[t] + F1 * Mv[EMBD + t] + F2 * Mv[2 * EMBD + t]) * invS
                + rest * bvrow[t]
                + a0 * (d2sh[0] * Mdv[t] + d2sh[1] * Mdv[EMBD + t] + bdvrow[t])
                + v0buf[b * EMBD + t];
        out[b * EMBD + t] = h;
    }
}

// ---------------------------------------------------------------------------
extern "C" void kernel_launch(void* const* d_in, const int* in_sizes, int n_in,
                              void* d_out, int out_size, void* d_ws, size_t ws_size,
                              hipStream_t stream) {
    const float* node_feats = (const float*)d_in[0];
    const float* state      = (const float*)d_in[1];
    const float* W_node     = (const float*)d_in[2];
    const float* b_node     = (const float*)d_in[3];
    const float* W_depot    = (const float*)d_in[4];
    const float* b_depot    = (const float*)d_in[5];
    const float* W_state    = (const float*)d_in[6];
    const float* b_state    = (const float*)d_in[7];
    const float* w_q        = (const float*)d_in[8];
    const float* w_k        = (const float*)d_in[9];
    const float* w_v        = (const float*)d_in[10];
    const int*   curr_id    = (const int*)d_in[11];
    const int*   next_id    = (const int*)d_in[12];
    const unsigned char* mask = (const unsigned char*)d_in[13];
    float* ws  = (float*)d_ws;
    float* out = (float*)d_out;

    fold_kernel<<<1, 128, 0, stream>>>(W_node, b_node, W_depot, b_depot, w_k, w_v,
                                       ws + OFF_FOLD);
    embed_kernel<<<BSZ, 128, 0, stream>>>(node_feats, state, W_node, b_node,
                                          W_depot, b_depot, W_state, b_state,
                                          curr_id, next_id, ws + OFF_QIN);
    // q  = Qin[256,384] @ w_q[384,128]
    gemm_wmma<<<32, 128, 0, stream>>>(ws + OFF_QIN, 384, w_q, EMBD, 384, ws + OFF_Q,  EMBD);
    // k0 = curr_emb[256,128] @ w_k_top[128,128]  (curr_emb = Qin cols 0..127)
    gemm_wmma<<<32, 128, 0, stream>>>(ws + OFF_QIN, 384, w_k, EMBD, 128, ws + OFF_K0, EMBD);
    // v0 = curr_emb[256,128] @ w_v_top[128,128]
    gemm_wmma<<<32, 128, 0, stream>>>(ws + OFF_QIN, 384, w_v, EMBD, 128, ws + OFF_V0, EMBD);

    attn_kernel<<<BSZ, 256, 0, stream>>>(node_feats, mask,
                                         ws + OFF_Q, ws + OFF_K0, ws + OFF_V0,
                                         ws + OFF_FOLD, out);
}